// ClusterLoss_25701084299447
// MI455X (gfx1250) — compile-verified
//
#include <hip/hip_runtime.h>

// CDNA5 / gfx1250. Memory-bound segmented-std reduction (256MB stream, AI~0.75):
//   rowsq  via V_WMMA_F32_16X16X4_F32 Gram diagonal  (c  = A x A^T, exact fp32;
//          B=A^T is register-identical to A in the 16x16x4 f32 layout)
//   rowsum via V_WMMA_F32_16X16X4_F32 with all-ones B (cs = A x 1; every column
//          of D equals the row sum, so this is layout-invariant)
//   class reduce via LDS ds_add_f32 atomics, block flush via global atomics.

typedef __attribute__((ext_vector_type(2))) float v2f;
typedef __attribute__((ext_vector_type(4))) float v4f;
typedef __attribute__((ext_vector_type(8))) float v8f;

#define NCLASS 64
#define DCOLS 128
#define TILE_ROWS 16
#define WAVES_PER_BLOCK 4
#define LDS_STRIDE 132  // 128 + 4 pad: bank = (4r + c) % 64 -> conflict-free b64 reads

__global__ void zero_ws_kernel(float* ws, int n) {
  int i = blockIdx.x * blockDim.x + threadIdx.x;
  if (i < n) ws[i] = 0.0f;
}

__global__ __launch_bounds__(WAVES_PER_BLOCK * 32)
void cluster_stats_kernel(const float* __restrict__ x,
                          const int* __restrict__ label,
                          float* __restrict__ gsum,
                          float* __restrict__ gsq,
                          float* __restrict__ gcnt,
                          int nrows, int ntiles) {
  __shared__ float tile[WAVES_PER_BLOCK][TILE_ROWS * LDS_STRIDE];
  __shared__ float csum[NCLASS], csq[NCLASS], ccnt[NCLASS];

  const int tid  = threadIdx.x;
  const int lane = tid & 31;   // wave32
  const int wave = tid >> 5;

  for (int i = tid; i < NCLASS; i += blockDim.x) {
    csum[i] = 0.f; csq[i] = 0.f; ccnt[i] = 0.f;
  }
  __syncthreads();

  const int t = blockIdx.x * WAVES_PER_BLOCK + wave;
  if (t < ntiles) {
    const int row0 = t * TILE_ROWS;
    float* lt = tile[wave];

    // Stage 16x128 fp32 tile: one full 512B row per instruction, coalesced,
    // non-temporal (the 256MB stream exceeds the 192MB L2 -> don't cache).
    if (row0 + TILE_ROWS <= nrows) {
#pragma unroll
      for (int r = 0; r < TILE_ROWS; ++r) {
        v4f v = __builtin_nontemporal_load(
            ((const v4f*)(x + (size_t)(row0 + r) * DCOLS)) + lane);
        ((v4f*)(lt + r * LDS_STRIDE))[lane] = v;
      }
    } else {  // ragged tail tile: zero-fill OOB rows (keeps WMMA contributions 0)
      for (int r = 0; r < TILE_ROWS; ++r) {
        const int grow = row0 + r;
        v4f v = {0.f, 0.f, 0.f, 0.f};
        if (grow < nrows)
          v = __builtin_nontemporal_load(
              ((const v4f*)(x + (size_t)grow * DCOLS)) + lane);
        ((v4f*)(lt + r * LDS_STRIDE))[lane] = v;
      }
    }
    // same-wave LDS RAW: compiler inserts s_wait_dscnt

    // WMMA f32 16x16x4 A layout: lane L<16 -> row L, K={0,1}; lane L+16 -> row L, K={2,3}.
    const int arow   = lane & 15;
    const int kshift = (lane >> 4) * 2;
    const float* rp  = lt + arow * LDS_STRIDE + kshift;

    const v2f ones = {1.f, 1.f};
    v8f c  = {};  // Gram accumulator: diag = row sum-of-squares
    v8f cs = {};  // A x 1 accumulator: every column = row sum
#pragma unroll
    for (int k0 = 0; k0 < DCOLS; k0 += 4) {
      v2f a = *(const v2f*)(rp + k0);
      c  = __builtin_amdgcn_wmma_f32_16x16x4_f32(
          false, a, false, a,    (short)0, c,  false, false);
      cs = __builtin_amdgcn_wmma_f32_16x16x4_f32(
          false, a, false, ones, (short)0, cs, false, false);
    }

    // Diagonal extraction (C/D layout): lanes 0..7 own rows 0..7 via idx=lane,
    // lanes 24..31 own rows 8..15 via idx=lane-24.
    const bool owner = (lane < 8) || (lane >= 24);
    const int  idx   = (lane < 8) ? lane : (lane - 24);
    float rowsq = 0.f, rowsum = 0.f;
#pragma unroll
    for (int j = 0; j < 8; ++j) {
      rowsq  = (idx == j) ? c[j]  : rowsq;
      rowsum = (idx == j) ? cs[j] : rowsum;
    }

    if (owner) {
      const int m = (lane < 8) ? lane : (lane - 16);  // row within tile
      const int grow = row0 + m;
      if (grow < nrows) {
        const int lab = label[grow];
        atomicAdd(&csum[lab], rowsum);   // ds_add_f32
        atomicAdd(&csq[lab],  rowsq);
        atomicAdd(&ccnt[lab], 1.0f);
      }
    }
  }
  __syncthreads();

  if (tid < NCLASS) {
    const float s = csum[tid], q = csq[tid], n = ccnt[tid];
    if (s != 0.f || q != 0.f || n != 0.f) {
      atomicAdd(&gsum[tid], s);
      atomicAdd(&gsq[tid],  q);
      atomicAdd(&gcnt[tid], n);
    }
  }
}

__global__ void finalize_kernel(const float* __restrict__ gsum,
                                const float* __restrict__ gsq,
                                const float* __restrict__ gcnt,
                                float* __restrict__ out) {
  __shared__ float partial[NCLASS];
  const int t = threadIdx.x;
  float v = 0.f;
  if (t < NCLASS) {
    const float n = gcnt[t] * (float)DCOLS;  // elements in class block
    if (n > 1.5f) {
      const float s = gsum[t], q = gsq[t];
      const float var = (q - s * s / n) / (n - 1.0f);
      v = sqrtf(fmaxf(var, 0.f));
    }
  }
  partial[t] = v;
  __syncthreads();
  if (t == 0) {
    float acc = 0.f;
    for (int i = 0; i < NCLASS; ++i) acc += partial[i];
    out[0] = acc;
  }
}

extern "C" void kernel_launch(void* const* d_in, const int* in_sizes, int n_in,
                              void* d_out, int out_size, void* d_ws, size_t ws_size,
                              hipStream_t stream) {
  const float* x     = (const float*)d_in[0];
  const int*   label = (const int*)d_in[1];
  const int nrows = in_sizes[1];  // N = label count

  float* gsum = (float*)d_ws;
  float* gsq  = gsum + NCLASS;
  float* gcnt = gsq + NCLASS;

  zero_ws_kernel<<<1, 3 * NCLASS, 0, stream>>>(gsum, 3 * NCLASS);

  const int ntiles = (nrows + TILE_ROWS - 1) / TILE_ROWS;
  const int blocks = (ntiles + WAVES_PER_BLOCK - 1) / WAVES_PER_BLOCK;
  cluster_stats_kernel<<<blocks, WAVES_PER_BLOCK * 32, 0, stream>>>(
      x, label, gsum, gsq, gcnt, nrows, ntiles);

  finalize_kernel<<<1, NCLASS, 0, stream>>>(gsum, gsq, gcnt, (float*)d_out);
}